// UnifiedObjectTransModule_17858474017289
// MI455X (gfx1250) — compile-verified
//
#include <hip/hip_runtime.h>
#include <math.h>
#include <stdint.h>

// Problem geometry (fixed by the reference setup_inputs()).
#define B_DIM   256
#define S_DIM   4096
#define T_TILE  128
#define NT      (S_DIM / T_TILE)      // 32 tiles per sequence
#define IN_FLOATS   (T_TILE * 21)     // cp(3)+hp(6)+rot(9)+gt(3) per frame = 2688
#define OUT_FLOATS  (T_TILE * 11)     // out3+fl+fr+fld+frd = 1408
#define IN_CHUNKS   (IN_FLOATS / 4)   // 672 x 16B
#define OUT_CHUNKS  (OUT_FLOATS / 4)  // 352 x 16B

// ---------------- CDNA5 async global->LDS helpers ----------------
#if defined(__has_builtin)
#if __has_builtin(__builtin_amdgcn_global_load_async_to_lds_b128)
#define HAVE_ASYNC_LOAD_BUILTIN 1
#endif
#if __has_builtin(__builtin_amdgcn_s_wait_asynccnt)
#define HAVE_WAIT_ASYNC_BUILTIN 1
#endif
#endif

// The builtin's parameter type (from the round-1 diagnostic):
//   addrspace(1) pointer to `int __attribute__((vector_size(16)))`
typedef int v4i_t __attribute__((vector_size(16)));
typedef __attribute__((address_space(1))) v4i_t* global_v4i_ptr;
typedef __attribute__((address_space(3))) v4i_t* lds_v4i_ptr;

__device__ __forceinline__ void async_load_b128(const void* g, void* l) {
#ifdef HAVE_ASYNC_LOAD_BUILTIN
  __builtin_amdgcn_global_load_async_to_lds_b128(
      (global_v4i_ptr)(uintptr_t)g,   // flat addr of global mem == global addr
      (lds_v4i_ptr)l,                 // addrspacecast generic -> local
      /*imm offset*/ 0, /*cpol*/ 0);
#else
  unsigned loff = (unsigned)(unsigned long long)
      (__attribute__((address_space(3))) void*)l;
  asm volatile("global_load_async_to_lds_b128 %0, %1, off"
               :: "v"(loff), "v"(g) : "memory");
#endif
}

#ifdef HAVE_WAIT_ASYNC_BUILTIN
#define WAIT_ASYNC_LE(n) __builtin_amdgcn_s_wait_asynccnt(n)
#else
#define WAIT_ASYNC_LE(n) asm volatile("s_wait_asynccnt %0" :: "i"(n) : "memory")
#endif

// ---------------- kernel ----------------
__global__ __launch_bounds__(256)
void fk_scan_kernel(const float* __restrict__ cp,   // [B,S,3]
                    const float* __restrict__ hp,   // [B,S,2,3]
                    const float* __restrict__ rot,  // [B,S,3,3]
                    const float* __restrict__ gt,   // [B,S,3]
                    float* __restrict__ out)        // out3|fl|fr|fld|frd flat
{
  __shared__ alignas(16) float s_in[2][IN_FLOATS];   // double-buffered input tiles
  __shared__ alignas(16) float s_out[2][OUT_FLOATS]; // double-buffered output tiles

  const int b   = blockIdx.x;
  const int tid = threadIdx.x;
  const size_t base = (size_t)b * S_DIM;
  const size_t M    = (size_t)B_DIM * S_DIM;

  float* o3g  = out;            // [B,S,3]
  float* flg  = out + 3 * M;    // [B,S]
  float* frg  = out + 4 * M;    // [B,S]
  float* fldg = out + 5 * M;    // [B,S,3]
  float* frdg = out + 8 * M;    // [B,S,3]

  // Issue async copies of one 128-frame tile into LDS buffer (k&1).
  // Every thread issues exactly 3 async ops -> every wave's ASYNCcnt += 3.
  auto issue_tile = [&](int k) {
    const int t0 = k * T_TILE;
    const float* cpt = cp  + (base + (size_t)t0) * 3;
    const float* hpt = hp  + (base + (size_t)t0) * 6;
    const float* rtt = rot + (base + (size_t)t0) * 9;
    const float* gtt = gt  + (base + (size_t)t0) * 3;
    float* lb = &s_in[k & 1][0];
#pragma unroll
    for (int j = 0; j < 3; ++j) {
      int c = tid + j * 256;
      if (c >= IN_CHUNKS) c = IN_CHUNKS - 1;   // uniform issue count; dup is benign
      const float* g;
      if (c < 96)       g = cpt + (size_t)c * 4;
      else if (c < 288) g = hpt + (size_t)(c - 96) * 4;
      else if (c < 576) g = rtt + (size_t)(c - 288) * 4;
      else              g = gtt + (size_t)(c - 576) * 4;
      async_load_b128(g, lb + (size_t)c * 4);
    }
  };

  // Scan state (lives in lane 0's registers; mirrors jax.lax.scan carry).
  int   cur = 0;
  bool  started = false, lprev = false, rprev = false;
  float blen = 0.f, bdx = 0.f, bdy = 0.f, bdz = 0.f;
  float cox = 0.f, coy = 0.f, coz = 0.f;

  issue_tile(0);

  for (int k = 0; k < NT; ++k) {
    if (k + 1 < NT) { issue_tile(k + 1); WAIT_ASYNC_LE(3); }
    else            { WAIT_ASYNC_LE(0); }
    __syncthreads();   // tile k input fully resident in LDS for all waves

    if (tid == 0) {
      const float* li  = &s_in[k & 1][0];
      const float* pcp = li;                 // [T,3]
      const float* php = li + T_TILE * 3;    // [T,6]
      const float* prt = li + T_TILE * 9;    // [T,9]  (offset 1152)
      const float* pgt = li + T_TILE * 18;   // [T,3]  (offset 2304)
      float* lo   = &s_out[k & 1][0];
      float* lo3  = lo;                // [T,3]
      float* lfl  = lo + 384;          // [T]
      float* lfr  = lo + 512;          // [T]
      float* lfld = lo + 640;          // [T,3]
      float* lfrd = lo + 1024;         // [T,3]
      const int t0 = k * T_TILE;

      for (int i = 0; i < T_TILE; ++i) {
        const bool nz = (t0 + i) != 0;       // frame 0 forced to no-contact
        const bool lc = nz && (pcp[i * 3 + 0] > 0.5f);
        const bool rc = nz && (pcp[i * 3 + 1] > 0.5f);
        const bool ls = lc && !lprev;
        const bool rs = rc && !rprev;
        lprev = lc; rprev = rc;

        const int  nc   = ls ? 1 : (rs ? 2 : 0);
        const bool hasc = (cur == 1) ? lc : rc;
        const bool hasa = (cur == 2) ? lc : rc;
        const bool cA   = nc > 0;
        const bool cB   = !cA && !hasc && hasa && (cur != 0);
        const bool cC   = !cA && !cB && !hasc && (cur != 0);
        const int  sw   = (cur == 2) ? 1 : 2;
        const int  curn = cA ? nc : (cB ? sw : (cC ? 0 : cur));
        const bool start = cA || cB;

        float hx, hy, hz;
        if (curn == 1) { hx = php[i*6+0]; hy = php[i*6+1]; hz = php[i*6+2]; }
        else           { hx = php[i*6+3]; hy = php[i*6+4]; hz = php[i*6+5]; }
        const float gx = pgt[i*3+0], gy = pgt[i*3+1], gz = pgt[i*3+2];
        const float r00 = prt[i*9+0], r01 = prt[i*9+1], r02 = prt[i*9+2];
        const float r10 = prt[i*9+3], r11 = prt[i*9+4], r12 = prt[i*9+5];
        const float r20 = prt[i*9+6], r21 = prt[i*9+7], r22 = prt[i*9+8];

        if (start) {
          const float iox = started ? cox : gx;
          const float ioy = started ? coy : gy;
          const float ioz = started ? coz : gz;
          const float vx = iox - hx, vy = ioy - hy, vz = ioz - hz;
          const float d2 = vx*vx + vy*vy + vz*vz;
          float ux, uy, uz;
          if (d2 > 1e-12f) {
            const float d = sqrtf(d2);
            const float inv = 1.0f / d;
            ux = vx * inv; uy = vy * inv; uz = vz * inv;
            blen = d;
          } else {
            ux = 0.f; uy = 0.f; uz = 1.f;
            blen = 0.1f;
          }
          bdx = r00*ux + r10*uy + r20*uz;   // R^T @ unit
          bdy = r01*ux + r11*uy + r21*uz;
          bdz = r02*ux + r12*uy + r22*uz;
          started = true;
        }
        cur = curn;

        float ox, oy, oz;
        float flv = 0.f, frv = 0.f;
        float fdx = 0.f, fdy = 0.f, fdz = 0.f;
        float gdx = 0.f, gdy = 0.f, gdz = 0.f;
        if (curn != 0) {
          const float px = hx + (r00*bdx + r01*bdy + r02*bdz) * blen;
          const float py = hy + (r10*bdx + r11*bdy + r12*bdz) * blen;
          const float pz = hz + (r20*bdx + r21*bdy + r22*bdz) * blen;
          ox = px; oy = py; oz = pz;
          cox = px; coy = py; coz = pz;
          if (curn == 1) { flv = blen; fdx = bdx; fdy = bdy; fdz = bdz; }
          else           { frv = blen; gdx = bdx; gdy = bdy; gdz = bdz; }
        } else { ox = gx; oy = gy; oz = gz; }

        lo3[i*3+0] = ox; lo3[i*3+1] = oy; lo3[i*3+2] = oz;
        lfl[i] = flv; lfr[i] = frv;
        lfld[i*3+0] = fdx; lfld[i*3+1] = fdy; lfld[i*3+2] = fdz;
        lfrd[i*3+0] = gdx; lfrd[i*3+1] = gdy; lfrd[i*3+2] = gdz;
      }
    }
    __syncthreads();   // outputs staged; safe for everyone to read LDS

    // Coalesced float4 stores of the staged tile (352 x 16B chunks).
    {
      const int t0 = k * T_TILE;
      const float* lob = &s_out[k & 1][0];
      float* o3t  = o3g  + (base + (size_t)t0) * 3;
      float* flt  = flg  +  base + (size_t)t0;
      float* frt  = frg  +  base + (size_t)t0;
      float* fldt = fldg + (base + (size_t)t0) * 3;
      float* frdt = frdg + (base + (size_t)t0) * 3;

      {
        const int c = tid;                       // chunks 0..255
        const float4 v = *(const float4*)(lob + (size_t)c * 4);
        float* gp;
        if (c < 96)       gp = o3t  + (size_t)c * 4;
        else if (c < 128) gp = flt  + (size_t)(c - 96) * 4;
        else if (c < 160) gp = frt  + (size_t)(c - 128) * 4;
        else              gp = fldt + (size_t)(c - 160) * 4;
        *(float4*)gp = v;
      }
      if (tid < OUT_CHUNKS - 256) {              // chunks 256..351 -> frd
        const int c = 256 + tid;
        const float4 v = *(const float4*)(lob + (size_t)c * 4);
        *(float4*)(frdt + (size_t)(c - 256) * 4) = v;
      }
    }
    // No extra barrier needed: LDS values were read into registers before the
    // stores issued, and buffer (k&1) isn't rewritten until tile k+2, which is
    // separated by the two barriers of tile k+1.
  }
}

// ---------------- host shim ----------------
extern "C" void kernel_launch(void* const* d_in, const int* in_sizes, int n_in,
                              void* d_out, int out_size, void* d_ws, size_t ws_size,
                              hipStream_t stream) {
  (void)in_sizes; (void)n_in; (void)out_size; (void)d_ws; (void)ws_size;
  const float* cp  = (const float*)d_in[0];   // pred_hand_contact_prob [B,S,3]
  const float* hp  = (const float*)d_in[1];   // pred_hand_positions    [B,S,2,3]
  const float* rot = (const float*)d_in[2];   // obj_rot_matrix         [B,S,3,3]
  const float* gt  = (const float*)d_in[3];   // gt_obj_trans           [B,S,3]
  float* out = (float*)d_out;
  fk_scan_kernel<<<dim3(B_DIM), dim3(256), 0, stream>>>(cp, hp, rot, gt, out);
}